// DisulfideEnergy_49443663511892
// MI455X (gfx1250) — compile-verified
//
#include <hip/hip_runtime.h>
#include <hip/hip_bf16.h>
#include <stdint.h>

// Reference constants
#define SG_ID    7
#define NCHAINS  4
#define NRES     500
#define PD_EPS   1e-6f

typedef float v4f __attribute__((ext_vector_type(4)));
typedef int   v4i __attribute__((ext_vector_type(4)));

// ---------------------------------------------------------------------------
// Kernel 0: zero resi_energy + atom_energy regions (float4 stores)
// ---------------------------------------------------------------------------
__global__ void zero_f4_kernel(v4f* __restrict__ p, int n4) {
    int i = blockIdx.x * blockDim.x + threadIdx.x;
    if (i < n4) p[i] = (v4f){0.f, 0.f, 0.f, 0.f};
}

// ---------------------------------------------------------------------------
// Kernel 1: per-pair sulfur flag + rare-path energy scatter.
// Pair indices are staged block-wide into LDS with the CDNA5 async
// global->LDS DMA path (ASYNCcnt), then consumed from LDS.
// ---------------------------------------------------------------------------
__global__ __launch_bounds__(256) void pair_kernel(
    const float* __restrict__ coords,       // [n_atoms,3]
    const int*   __restrict__ desc,         // [n_atoms,4] at_name,resnum,batch,chain
    const int*   __restrict__ pairs,        // [n_pairs,2]
    const int*   __restrict__ amask,        // [n_atoms,4] bool-as-int
    float*       __restrict__ atom_energy,  // [n_atoms,4]
    float*       __restrict__ sulfur_out,   // [n_pairs]
    int n_pairs)
{
    __shared__ int2 spairs[256];

    const int  tid   = threadIdx.x;
    const long long p = (long long)blockIdx.x * 256 + tid;
    const bool valid = (p < (long long)n_pairs);
    const long long pc = valid ? p : (long long)n_pairs - 1;   // clamp tail lanes

    // ---- async stage this block's 256 pairs (2 KB) into LDS -------------
    // LDS destination comes from the first VGPR; global address is the SGPR base plus a VGPR byte offset
    unsigned lds_off = (unsigned)(uintptr_t)(&spairs[tid]);    // low 32 bits of
                                                               // generic ptr == LDS offset
    unsigned gbyte   = (unsigned)(pc * 8ll);                   // <= 32MB, fits u32
    asm volatile("global_load_async_to_lds_b64 %0, %1, %2"
                 :: "v"(lds_off), "v"(gbyte), "s"(pairs)
                 : "memory");
    asm volatile("s_wait_asynccnt 0x0" ::: "memory");
    __syncthreads();

    const int2 pr = spairs[tid];
    const int i = pr.x;
    const int j = pr.y;

    // Hot path: two 4B gathers from the 8MB (L2-resident) description table.
    const int ni = desc[4 * i + 0];
    const int nj = desc[4 * j + 0];
    const bool sulfur = (ni == SG_ID) & (nj == SG_ID);

    if (valid) {
        // write-once stream: keep out of near caches
        __builtin_nontemporal_store(sulfur ? 1.0f : 0.0f, &sulfur_out[p]);
    }

    // Rare path (~1/1600 of lanes): whole waves usually s_cbranch_execz over this.
    if (sulfur & valid) {
        const float dx = coords[3 * i + 0] - coords[3 * j + 0] + PD_EPS;
        const float dy = coords[3 * i + 1] - coords[3 * j + 1] + PD_EPS;
        const float dz = coords[3 * i + 2] - coords[3 * j + 2] + PD_EPS;
        const float dist = sqrtf(dx * dx + dy * dy + dz * dz);

        const int ri = desc[4 * i + 1];
        const int rj = desc[4 * j + 1];
        const float rd = fabsf((float)(ri - rj));
        const float rd_safe = (rd > 0.f) ? rd : 1.0f;

        // energy = -0.001*298*(2.1 + 2.9823825*ln(rd_safe)) + 5*|dist-2.04|
        const float energy = -0.298f * (2.1f + 2.9823825f * logf(rd_safe))
                           + 5.0f * fabsf(dist - 2.04f);
        const float net = 0.5f * energy;

        const v4i mi = *(const v4i*)(amask + 4ll * i);
        const v4i mj = *(const v4i*)(amask + 4ll * j);

        if (mi.x && mj.x) { atomicAdd(&atom_energy[4ll*i+0], net); atomicAdd(&atom_energy[4ll*j+0], net); }
        if (mi.y && mj.y) { atomicAdd(&atom_energy[4ll*i+1], net); atomicAdd(&atom_energy[4ll*j+1], net); }
        if (mi.z && mj.z) { atomicAdd(&atom_energy[4ll*i+2], net); atomicAdd(&atom_energy[4ll*j+2], net); }
        if (mi.w && mj.w) { atomicAdd(&atom_energy[4ll*i+3], net); atomicAdd(&atom_energy[4ll*j+3], net); }
    }
}

// ---------------------------------------------------------------------------
// Kernel 2: atom -> residue segment sum. Zero-skip: almost every atom has
// all-zero energy, so the common case is one NT float4 load + exit.
// ---------------------------------------------------------------------------
__global__ __launch_bounds__(256) void residue_kernel(
    const float* __restrict__ atom_energy,  // [n_atoms,4]
    const int*   __restrict__ desc,         // [n_atoms,4]
    float*       __restrict__ resi,         // [NB,NCHAINS,NRES,4] flat
    int n_atoms)
{
    int a = blockIdx.x * blockDim.x + threadIdx.x;
    if (a >= n_atoms) return;

    const v4f e = __builtin_nontemporal_load((const v4f*)(atom_energy + 4ll * a));
    if (e.x == 0.f && e.y == 0.f && e.z == 0.f && e.w == 0.f) return;

    const v4i d = *(const v4i*)(desc + 4ll * a);   // at_name,resnum,batch,chain
    const int flat = (d.z * NCHAINS + d.w) * NRES + d.y;

    if (e.x != 0.f) atomicAdd(&resi[4 * flat + 0], e.x);
    if (e.y != 0.f) atomicAdd(&resi[4 * flat + 1], e.y);
    if (e.z != 0.f) atomicAdd(&resi[4 * flat + 2], e.z);
    if (e.w != 0.f) atomicAdd(&resi[4 * flat + 3], e.w);
}

// ---------------------------------------------------------------------------
extern "C" void kernel_launch(void* const* d_in, const int* in_sizes, int n_in,
                              void* d_out, int out_size, void* d_ws, size_t ws_size,
                              hipStream_t stream) {
    const float* coords = (const float*)d_in[0];
    const int*   desc   = (const int*)d_in[1];
    const int*   pairs  = (const int*)d_in[2];
    const int*   amask  = (const int*)d_in[3];
    // d_in[4] = facc: unused by the reference computation.

    const int n_atoms = in_sizes[0] / 3;   // 500000
    const int n_pairs = in_sizes[2] / 2;   // 4000000
    const int n_resi  = out_size - 4 * n_atoms - n_pairs;   // 8*4*500*4 = 64000

    float* resi   = (float*)d_out;
    float* atomE  = resi + n_resi;
    float* sulfur = atomE + 4ll * n_atoms;

    // Kernel 0: zero resi_energy + atom_energy (both multiples of 4 floats)
    const int n4 = (n_resi + 4 * n_atoms) / 4;
    zero_f4_kernel<<<(n4 + 255) / 256, 256, 0, stream>>>((v4f*)d_out, n4);

    // Kernel 1: pairs
    pair_kernel<<<(n_pairs + 255) / 256, 256, 0, stream>>>(
        coords, desc, pairs, amask, atomE, sulfur, n_pairs);

    // Kernel 2: segment sum into residues
    residue_kernel<<<(n_atoms + 255) / 256, 256, 0, stream>>>(
        atomE, desc, resi, n_atoms);
}